// SysMaxOfAtoms_80719615361742
// MI455X (gfx1250) — compile-verified
//
#include <hip/hip_runtime.h>
#include <stdint.h>

// Segment-max (sorted segments): features[N_ATOMS][256] f32 -> out[N_MOL][256] f32.
// HBM-bound: ~1.02 GB read @ 23.3 TB/s => ~45 us roofline. One block per
// molecule, 64 lanes (2 wave32), lane t owns features [4t,4t+4).
// Streaming path uses the CDNA5 async copy engine:
//   global_load_async_to_lds_b128  (ASYNCcnt-tracked DMA into an 8-slot LDS ring)
//   s_wait_asynccnt                (throttle: oldest slot complete)
//   ds_load_b128                   (consume lane-private slice, no barriers)
// Async loads complete in-order per wave, so wait<=NBUF-1 releases the oldest.

#define NBUF 8            // ring slots (power of two)
#define LANES 64          // feat/4 lanes; 1 row = 64 * 16B = 1024B

__device__ __forceinline__ float4 f4max(float4 a, float4 b) {
    return make_float4(fmaxf(a.x, b.x), fmaxf(a.y, b.y),
                       fmaxf(a.z, b.z), fmaxf(a.w, b.w));
}

__device__ __forceinline__ int lower_bound_i32(const int* __restrict__ idx,
                                               int lo, int hi, int key) {
    while (lo < hi) {
        int mid = (lo + hi) >> 1;
        if (idx[mid] < key) lo = mid + 1; else hi = mid;
    }
    return lo;
}

// Issue one lane's 16B async copy: from the global address base plus goff to the LDS address in ldsaddr.
// s_wait_dscnt 0 first: the previous ds_load from this slot must have fully
// completed before the DMA may overwrite it (ASYNC and DS ops are unordered).
__device__ __forceinline__ void async_issue(const float* base, int goff,
                                            uint32_t ldsaddr) {
    asm volatile("s_wait_dscnt 0x0\n\t"
                 "global_load_async_to_lds_b128 %0, %1, %2\n\t"
                 :: "v"(ldsaddr), "v"(goff), "s"(base)
                 : "memory");
}

__device__ __forceinline__ void wait_async_oldest() {
    asm volatile("s_wait_asynccnt 0x7" ::: "memory");   // <= NBUF-1 outstanding
}
__device__ __forceinline__ void wait_async_all() {
    asm volatile("s_wait_asynccnt 0x0" ::: "memory");
}

__global__ __launch_bounds__(LANES) void SysMaxOfAtoms_segmax_async_kernel(
    const float* __restrict__ features,
    const int*   __restrict__ mol_index,
    float*       __restrict__ out,
    int n_atoms, int feat)
{
    __shared__ float4 buf[NBUF][LANES];   // 8 KB ring

    const int mol = blockIdx.x;
    const int tid = threadIdx.x;
    const int rowbytes = feat * 4;        // 1024

    // Uniform binary search (compiler scalarizes to s_load + SALU loop).
    const int start = lower_bound_i32(mol_index, 0, n_atoms, mol);
    const int end   = lower_bound_i32(mol_index, start, n_atoms, mol + 1);
    const int nrows = end - start;

    // Low 32 bits of the generic address of a __shared__ object == DS byte
    // offset within the wave's LDS allocation (aperture lives in the high half).
    const uint32_t lds_lane = (uint32_t)(uintptr_t)(&buf[0][tid]);

    const float NEG_INF = -__builtin_inff();
    float4 m = make_float4(NEG_INF, NEG_INF, NEG_INF, NEG_INF);

    // ---- prologue: fill up to NBUF slots -------------------------------
    int issued = 0;
    const int pro = (nrows < NBUF) ? nrows : NBUF;
    for (; issued < pro; ++issued) {
        async_issue(features,
                    (start + issued) * rowbytes + tid * 16,
                    lds_lane + (uint32_t)((issued & (NBUF - 1)) * (LANES * 16)));
    }

    // ---- steady state: consume row r, refill slot with row issued ------
    int r = 0;
    for (; issued < nrows; ++issued, ++r) {
        wait_async_oldest();                       // row r has landed
        float4 v = buf[r & (NBUF - 1)][tid];       // ds_load_b128
        m = f4max(m, v);
        async_issue(features,
                    (start + issued) * rowbytes + tid * 16,
                    lds_lane + (uint32_t)((issued & (NBUF - 1)) * (LANES * 16)));
    }

    // ---- drain ---------------------------------------------------------
    wait_async_all();
    for (; r < nrows; ++r) {
        float4 v = buf[r & (NBUF - 1)][tid];
        m = f4max(m, v);
    }

    // Reference maps -inf (empty segments never touched) to 0, elementwise.
    m.x = (m.x == NEG_INF) ? 0.0f : m.x;
    m.y = (m.y == NEG_INF) ? 0.0f : m.y;
    m.z = (m.z == NEG_INF) ? 0.0f : m.z;
    m.w = (m.w == NEG_INF) ? 0.0f : m.w;

    float4* orow = reinterpret_cast<float4*>(out)
                 + (size_t)mol * (size_t)(feat / 4) + (size_t)tid;
    *orow = m;   // one global_store_b128 per lane, written exactly once
}

extern "C" void kernel_launch(void* const* d_in, const int* in_sizes, int n_in,
                              void* d_out, int out_size, void* d_ws, size_t ws_size,
                              hipStream_t stream) {
    const float* features  = (const float*)d_in[0];
    const int*   mol_index = (const int*)d_in[1];
    // d_in[2] (n_molecules) lives on device; derive shapes host-side:
    const int n_atoms = in_sizes[1];
    const int feat    = in_sizes[0] / n_atoms;   // 256
    const int n_mol   = out_size / feat;         // 20000

    dim3 grid(n_mol);
    dim3 block(LANES);                           // feat/4 == 64 lanes
    SysMaxOfAtoms_segmax_async_kernel<<<grid, block, 0, stream>>>(
        features, mol_index, (float*)d_out, n_atoms, feat);
}